// NGramRepeatBlock_31086973288472
// MI455X (gfx1250) — compile-verified
//
#include <hip/hip_runtime.h>
#include <hip/hip_bf16.h>

// ---------------------------------------------------------------------------
// NGramRepeatBlock for MI455X (gfx1250).
// Memory-bound masked copy: 131 MB in + 131 MB out => ~11.3 us floor @23.3TB/s.
// Strategy: (1) full-bandwidth non-temporal B128 streaming copy of lprobs
//               (8 loads in flight per wave, no per-iter waits),
//           (2) tiny per-row ban pass that scatters -inf into the output.
// ---------------------------------------------------------------------------

typedef __attribute__((ext_vector_type(4))) float f4;

#define COPY_TPB 256
#define COPY_F4_PER_THREAD 8   // 256 threads * 8 f4 = 2048 f4 = 32 KB per block

// Exact-tile copy: no bounds checks so the compiler can issue all 8
// global_load_b128 NT back-to-back, one s_wait_loadcnt, then 8 stores.
__global__ __launch_bounds__(COPY_TPB) void ngram_copy_nt_full(
    const f4* __restrict__ src, f4* __restrict__ dst) {
  const long base =
      (long)blockIdx.x * (COPY_TPB * COPY_F4_PER_THREAD) + threadIdx.x;
  // Pull the stream ahead into L2 (gfx1250 global_prefetch_b8).
  __builtin_prefetch(
      (const void*)(src + base + 2L * COPY_TPB * COPY_F4_PER_THREAD), 0, 0);
  f4 v[COPY_F4_PER_THREAD];
#pragma unroll
  for (int k = 0; k < COPY_F4_PER_THREAD; ++k) {
    v[k] = __builtin_nontemporal_load(&src[base + (long)k * COPY_TPB]);
  }
#pragma unroll
  for (int k = 0; k < COPY_F4_PER_THREAD; ++k) {
    __builtin_nontemporal_store(v[k], &dst[base + (long)k * COPY_TPB]);
  }
}

// Guarded tail copy (grid-stride); only launched for the (normally zero)
// remainder that doesn't fill a 2048-f4 block.
__global__ __launch_bounds__(COPY_TPB) void ngram_copy_nt_tail(
    const f4* __restrict__ src, f4* __restrict__ dst, long start, long n4) {
  for (long idx = start + blockIdx.x * (long)blockDim.x + threadIdx.x;
       idx < n4; idx += (long)gridDim.x * blockDim.x) {
    f4 v = __builtin_nontemporal_load(&src[idx]);
    __builtin_nontemporal_store(v, &dst[idx]);
  }
}

// One block per row. NGRAM == 3: prefix = (tok[step-1], tok[step]);
// window i in [0, L-2) matches if (tok[i],tok[i+1]) == prefix -> ban tok[i+2].
#define BAN_TPB 256

__global__ __launch_bounds__(BAN_TPB) void ngram_ban(
    const int* __restrict__ tokens, float* __restrict__ out,
    int L, int V, int step) {
  extern __shared__ int tok[];
  const int row = blockIdx.x;
  const int* trow = tokens + (size_t)row * (size_t)L;
  for (int i = threadIdx.x; i < L; i += BAN_TPB) tok[i] = trow[i];
  __syncthreads();

  const int p0 = tok[step - 1];
  const int p1 = tok[step];
  const int W = L - 2;  // number of 3-gram windows
  float* orow = out + (size_t)row * (size_t)V;
  const float ninf = -__builtin_huge_valf();

  for (int i = threadIdx.x; i < W; i += BAN_TPB) {
    if (tok[i] == p0 && tok[i + 1] == p1) {
      const int banned = tok[i + 2];
      if ((unsigned)banned < (unsigned)V) {
        orow[banned] = ninf;  // racing lanes all write -inf: benign
      }
    }
  }
}

// ---------------------------------------------------------------------------
// TDM probe (never launched): this toolchain exposes the 6-arg clang-23
// tensor_load_to_lds builtin:
//   (u32x4 group0, i32x8 group1, i32x4 group2, i32x4 group3, i32x8 group4,
//    i32 cpol)
// Compiling it in confirms the tensor op + s_wait_tensorcnt lower for gfx1250.
// ---------------------------------------------------------------------------
typedef __attribute__((ext_vector_type(4))) unsigned int u32x4;
typedef __attribute__((ext_vector_type(8))) int i32x8;
typedef __attribute__((ext_vector_type(4))) int i32x4;

#if defined(__gfx1250__) && __has_builtin(__builtin_amdgcn_tensor_load_to_lds)
__global__ void tdm_arity_probe() {
  u32x4 g0 = {0u, 0u, 0u, 0u};
  i32x8 g1 = {0, 0, 0, 0, 0, 0, 0, 0};
  i32x4 g2 = {0, 0, 0, 0};
  i32x4 g3 = {0, 0, 0, 0};
  i32x8 g4 = {0, 0, 0, 0, 0, 0, 0, 0};
  __builtin_amdgcn_tensor_load_to_lds(g0, g1, g2, g3, g4, 0);
  __builtin_amdgcn_s_wait_tensorcnt(0);
}
#endif

// ---------------------------------------------------------------------------

extern "C" void kernel_launch(void* const* d_in, const int* in_sizes, int n_in,
                              void* d_out, int out_size, void* d_ws,
                              size_t ws_size, hipStream_t stream) {
  const int* tokens = (const int*)d_in[0];
  const float* lprobs = (const float*)d_in[1];
  float* out = (float*)d_out;

  // Reference fixes bsz=32, beam=8 -> R=256; derive the rest from in_sizes.
  const int R = 32 * 8;
  const int L = in_sizes[0] / R;  // 512
  const int V = in_sizes[1] / R;  // 128000
  const int step = L - 1;         // 511

  // 1) Streaming copy of lprobs -> out (full bandwidth, NT hints).
  const long n4 = (long)out_size / 4;  // out_size is element count, /4 = f4s
  const long perBlock = (long)COPY_TPB * COPY_F4_PER_THREAD;
  const long fullBlocks = n4 / perBlock;          // 4000 for these shapes
  if (fullBlocks > 0) {
    ngram_copy_nt_full<<<(int)fullBlocks, COPY_TPB, 0, stream>>>(
        (const f4*)lprobs, (f4*)out);
  }
  const long tailStart = fullBlocks * perBlock;
  if (tailStart < n4) {
    ngram_copy_nt_tail<<<8, COPY_TPB, 0, stream>>>((const f4*)lprobs, (f4*)out,
                                                   tailStart, n4);
  }

  // 2) Ban pass: scatter -inf at repeated-ngram continuations (after the copy
  //    by stream ordering). Skip if prefix would start before position 0.
  if (step - 1 >= 0) {
    ngram_ban<<<R, BAN_TPB, L * (int)sizeof(int), stream>>>(tokens, out, L, V,
                                                            step);
  }
}